// GraphTransformerGather_36326833389872
// MI455X (gfx1250) — compile-verified
//
#include <hip/hip_runtime.h>

#define NN   3072
#define EE   98304
#define DD   128
#define HH   8
#define EDIM 16
#define LL   3
#define FFD  256
#define DHD  16
#define SCALE 0.25f   // DH^-0.5
#define MCH  8        // attention m-dimension split (occupancy)
#define MROWS (NN / MCH)   // 384 rows per chunk = 24 m-tiles

typedef float v2f __attribute__((ext_vector_type(2)));
typedef float v8f __attribute__((ext_vector_type(8)));

// ---------------------------------------------------------------------------
// order-preserving float <-> u32 map for atomicMax-based segment max
// ---------------------------------------------------------------------------
__device__ __forceinline__ unsigned fmap(float f) {
    unsigned u = __float_as_uint(f);
    return (u & 0x80000000u) ? ~u : (u | 0x80000000u);
}
__device__ __forceinline__ float funmap(unsigned m) {
    if (m == 0u) return 0.0f;                 // empty segment -> 0 (ref isfinite rule)
    unsigned u = (m & 0x80000000u) ? (m & 0x7fffffffu) : ~m;
    float f = __uint_as_float(u);
    if (!(fabsf(f) < 1e37f)) return 0.0f;     // non-finite -> 0
    return f;
}

// ---------------------------------------------------------------------------
// edge bias: a[e,h] = edge_feats[e,:] @ W_edge[:,h] + b_edge[h]
// ---------------------------------------------------------------------------
__global__ void edge_bias_kernel(const float* __restrict__ ef,
                                 const float* __restrict__ We,
                                 const float* __restrict__ be,
                                 float* __restrict__ a) {
    int i = blockIdx.x * blockDim.x + threadIdx.x;
    if (i >= EE * HH) return;
    int e = i >> 3, hh = i & 7;
    const float* row = ef + e * EDIM;
    float acc = be[hh];
#pragma unroll
    for (int j = 0; j < EDIM; ++j) acc += row[j] * We[j * HH + hh];
    a[i] = acc;
}

// ---------------------------------------------------------------------------
// init segment-max buffer to mapped "-inf" (0)
// ---------------------------------------------------------------------------
__global__ void smax_init_kernel(unsigned* __restrict__ smax) {
    int i = blockIdx.x * blockDim.x + threadIdx.x;
    if (i < NN * HH) smax[i] = 0u;
}

// ---------------------------------------------------------------------------
// per-edge score + atomic segment max over dst (max is order-independent ->
// deterministic): s = scale * q[src[e],h,:].k[dst[e],h,:] + a[e,h]
// ---------------------------------------------------------------------------
__global__ void edge_smax_kernel(const float* __restrict__ q,
                                 const float* __restrict__ k,
                                 const float* __restrict__ a,
                                 const int* __restrict__ src,
                                 const int* __restrict__ dst,
                                 unsigned* __restrict__ smax) {
    int i = blockIdx.x * blockDim.x + threadIdx.x;
    if (i >= EE * HH) return;
    int e = i >> 3, hh = i & 7;
    int s = src[e], d = dst[e];
    const float* qr = q + (size_t)s * DD + hh * DHD;
    const float* kr = k + (size_t)d * DD + hh * DHD;
    float acc = 0.0f;
#pragma unroll
    for (int j = 0; j < DHD; ++j) acc += qr[j] * kr[j];
    float val = acc * SCALE + a[i];
    atomicMax(smax + (size_t)d * HH + hh, fmap(val));
}

// ---------------------------------------------------------------------------
// Generic WMMA fp32 GEMM: C = A(MxK) @ B(KxNc) + bias [+ resid] [ReLU]
// One wave computes one 16x16 C tile with K/4 chained v_wmma_f32_16x16x4_f32.
// ---------------------------------------------------------------------------
__global__ __launch_bounds__(128)
void gemm16_kernel(const float* __restrict__ A, const float* __restrict__ B,
                   const float* __restrict__ bias, const float* __restrict__ resid,
                   float* __restrict__ C, int M, int K, int Nc, int relu) {
    int lane = threadIdx.x & 31;
    int wave = threadIdx.x >> 5;
    int tile = blockIdx.x * 4 + wave;
    int ntn  = Nc >> 4;
    int tm = tile / ntn, tn = tile % ntn;
    if (tm >= (M >> 4)) return;          // uniform per wave -> EXEC stays all-ones

    int r  = lane & 15;
    int kh = (lane >> 4) << 1;           // 0 or 2

    const float* Ap = A + (size_t)(tm * 16 + r) * K + kh;
    const float* Bp = B + (size_t)kh * Nc + tn * 16 + r;

    v8f acc = {};
    for (int kk = 0; kk < K; kk += 4) {
        __builtin_prefetch(Ap + kk + 64, 0, 3);   // near-cache (WGP) prefetch
        v2f a; a.x = Ap[kk];                 a.y = Ap[kk + 1];
        v2f b; b.x = Bp[(size_t)kk * Nc];    b.y = Bp[(size_t)(kk + 1) * Nc];
        acc = __builtin_amdgcn_wmma_f32_16x16x4_f32(false, a, false, b,
                                                    (short)0, acc, false, false);
    }

    int cn = tn * 16 + r;
    int rb = tm * 16 + ((lane >> 4) << 3);
    float bv = bias ? bias[cn] : 0.0f;
#pragma unroll
    for (int j = 0; j < 8; ++j) {
        float v = acc[j] + bv;
        if (resid) v += resid[(size_t)(rb + j) * Nc + cn];
        if (relu)  v = fmaxf(v, 0.0f);
        C[(size_t)(rb + j) * Nc + cn] = v;
    }
}

// ---------------------------------------------------------------------------
// Fused attention, m-split for occupancy. One wave owns a (16-row n-tile,
// m-chunk) pair: blockIdx.x covers n-tiles (4 waves/block), blockIdx.y = chunk.
// Per 16-row m-tile: 8 heads x 4 WMMA (scores, K=DH=16), in-lane softmax over
// the HEAD axis (element-local!), restage attn tile via wave-private LDS
// (C-layout -> A-layout), 8 heads x 4 WMMA into per-chunk output accumulators.
// Partials summed later by a deterministic reduce kernel (no float atomics).
// ---------------------------------------------------------------------------
__global__ __launch_bounds__(128)
void attn_kernel(const float* __restrict__ q, const float* __restrict__ k,
                 const float* __restrict__ v, const unsigned* __restrict__ smax,
                 float* __restrict__ ctx_part) {
    __shared__ float lds[4][16][17];     // wave-private 16x16 restage slab (padded)

    int lane  = threadIdx.x & 31;
    int wave  = threadIdx.x >> 5;
    int tile  = blockIdx.x * 4 + wave;   // 0..191 n-tile
    int chunk = blockIdx.y;              // 0..MCH-1
    int n0    = tile * 16;
    int mlo   = chunk * MROWS;
    int mhi   = mlo + MROWS;
    int r     = lane & 15;
    int kh    = (lane >> 4) << 1;        // 0 or 2
    int off8  = (lane >> 4) << 3;        // 0 or 8

    // q fragments in A-layout, all heads, all K-chunks (kept in VGPRs)
    v2f qf[HH][4];
#pragma unroll
    for (int h = 0; h < HH; ++h)
#pragma unroll
        for (int c = 0; c < 4; ++c) {
            const float* p = q + (size_t)(n0 + r) * DD + h * DHD + 4 * c + kh;
            qf[h][c].x = p[0];
            qf[h][c].y = p[1];
        }

    // per-(row,head) segment max, aligned with C-layout slots
    float smr[HH][8];
#pragma unroll
    for (int h = 0; h < HH; ++h)
#pragma unroll
        for (int j = 0; j < 8; ++j)
            smr[h][j] = funmap(smax[(size_t)(n0 + off8 + j) * HH + h]);

    v8f outacc[HH];
#pragma unroll
    for (int h = 0; h < HH; ++h) outacc[h] = (v8f){};

    for (int m0 = mlo; m0 < mhi; m0 += 16) {
        // ---- scores: sc[h] = q_h (16x16) @ k_h^T (16x16), K=16 -> 4 WMMAs ----
        v8f sc[HH];
#pragma unroll
        for (int h = 0; h < HH; ++h) {
            sc[h] = (v8f){};
            const float* kp = k + (size_t)(m0 + r) * DD + h * DHD + kh;
#pragma unroll
            for (int c = 0; c < 4; ++c) {
                v2f b; b.x = kp[4 * c]; b.y = kp[4 * c + 1];
                sc[h] = __builtin_amdgcn_wmma_f32_16x16x4_f32(
                    false, qf[h][c], false, b, (short)0, sc[h], false, false);
            }
        }
        // ---- softmax over the HEAD axis: purely element-wise per lane ----
#pragma unroll
        for (int j = 0; j < 8; ++j) {
            float t[HH], mx = -1e30f;
#pragma unroll
            for (int h = 0; h < HH; ++h) {
                t[h] = sc[h][j] * SCALE - smr[h][j];
                mx = fmaxf(mx, t[h]);
            }
            float s = 0.0f;
#pragma unroll
            for (int h = 0; h < HH; ++h) { t[h] = __expf(t[h] - mx); s += t[h]; }
            float inv = 1.0f / s;
#pragma unroll
            for (int h = 0; h < HH; ++h) sc[h][j] = t[h] * inv;
        }
        // ---- out += attn_h (16x16) @ v_h (16x16); restage attn via LDS ----
#pragma unroll
        for (int h = 0; h < HH; ++h) {
#pragma unroll
            for (int j = 0; j < 8; ++j) lds[wave][off8 + j][r] = sc[h][j];
            // same-wave LDS RAW: DS ops are in-order per wave
            const float* vp = v + (size_t)m0 * DD + h * DHD + r;
#pragma unroll
            for (int c = 0; c < 4; ++c) {
                v2f a; a.x = lds[wave][r][4 * c + kh];
                       a.y = lds[wave][r][4 * c + kh + 1];
                v2f b; b.x = vp[(size_t)(4 * c + kh) * DD];
                       b.y = vp[(size_t)(4 * c + kh + 1) * DD];
                outacc[h] = __builtin_amdgcn_wmma_f32_16x16x4_f32(
                    false, a, false, b, (short)0, outacc[h], false, false);
            }
        }
    }

    float* outp = ctx_part + (size_t)chunk * NN * DD;
#pragma unroll
    for (int h = 0; h < HH; ++h)
#pragma unroll
        for (int j = 0; j < 8; ++j)
            outp[(size_t)(n0 + off8 + j) * DD + h * DHD + r] = outacc[h][j];
}

// ---------------------------------------------------------------------------
// Deterministic reduction of the MCH partial ctx slabs (fixed summation order)
// ---------------------------------------------------------------------------
__global__ void ctx_reduce_kernel(const float* __restrict__ part,
                                  float* __restrict__ ctx) {
    int i = blockIdx.x * blockDim.x + threadIdx.x;
    if (i >= NN * DD) return;
    float s = 0.0f;
#pragma unroll
    for (int c = 0; c < MCH; ++c) s += part[(size_t)c * NN * DD + i];
    ctx[i] = s;
}

// ---------------------------------------------------------------------------
// LayerNorm: out = LN(hin + z) * gamma + beta ; one 128-thread block per row
// ---------------------------------------------------------------------------
__global__ void ln_kernel(const float* __restrict__ hin, const float* __restrict__ z,
                          const float* __restrict__ gamma, const float* __restrict__ beta,
                          float* __restrict__ out) {
    __shared__ float red[128];
    int row = blockIdx.x, t = threadIdx.x;
    float x = hin[(size_t)row * DD + t] + z[(size_t)row * DD + t];
    red[t] = x; __syncthreads();
    for (int s = 64; s > 0; s >>= 1) { if (t < s) red[t] += red[t + s]; __syncthreads(); }
    float mean = red[0] * (1.0f / DD);
    __syncthreads();
    float d = x - mean;
    red[t] = d * d; __syncthreads();
    for (int s = 64; s > 0; s >>= 1) { if (t < s) red[t] += red[t + s]; __syncthreads(); }
    float var = red[0] * (1.0f / DD);
    out[(size_t)row * DD + t] = d * rsqrtf(var + 1e-5f) * gamma[t] + beta[t];
}

// ---------------------------------------------------------------------------
extern "C" void kernel_launch(void* const* d_in, const int* in_sizes, int n_in,
                              void* d_out, int out_size, void* d_ws, size_t ws_size,
                              hipStream_t stream) {
    const float* node = (const float*)d_in[0];
    const float* ef   = (const float*)d_in[1];
    const float* We   = (const float*)d_in[2];
    const float* be   = (const float*)d_in[3];
    const float* Wq   = (const float*)d_in[4];
    const float* bq   = (const float*)d_in[5];
    const float* Wk   = (const float*)d_in[6];
    const float* bk   = (const float*)d_in[7];
    const float* Wv   = (const float*)d_in[8];
    const float* bv   = (const float*)d_in[9];
    const float* Wo   = (const float*)d_in[10];
    const float* bo   = (const float*)d_in[11];
    const float* W1   = (const float*)d_in[12];
    const float* b1   = (const float*)d_in[13];
    const float* W2   = (const float*)d_in[14];
    const float* b2   = (const float*)d_in[15];
    const float* gam  = (const float*)d_in[16];
    const float* bet  = (const float*)d_in[17];
    const int*   src  = (const int*)d_in[18];
    const int*   dst  = (const int*)d_in[19];
    float* out = (float*)d_out;

    float* ws = (float*)d_ws;
    size_t off = 0;
    float* a_edge   = ws + off; off += (size_t)EE * HH;
    float* q        = ws + off; off += (size_t)NN * DD;
    float* k        = ws + off; off += (size_t)NN * DD;
    float* v        = ws + off; off += (size_t)NN * DD;
    float* ctx      = ws + off; off += (size_t)NN * DD;
    float* h2       = ws + off; off += (size_t)NN * DD;
    float* ffn1     = ws + off; off += (size_t)NN * FFD;
    float* zbuf     = ws + off; off += (size_t)NN * DD;
    float* hb0      = ws + off; off += (size_t)NN * DD;
    float* hb1      = ws + off; off += (size_t)NN * DD;
    float* ctx_part = ws + off; off += (size_t)MCH * NN * DD;
    unsigned* smax  = (unsigned*)(ws + off); off += (size_t)NN * HH;

    edge_bias_kernel<<<(EE * HH + 255) / 256, 256, 0, stream>>>(ef, We, be, a_edge);

    const int gemmD  = (NN / 16) * (DD / 16) / 4;   // 384 blocks, 4 waves each
    const int gemmFF = (NN / 16) * (FFD / 16) / 4;  // 768 blocks

    for (int l = 0; l < LL; ++l) {
        const float* hin = (l == 0) ? node : ((l == 1) ? hb0 : hb1);
        float* hout      = (l == 2) ? out  : ((l == 0) ? hb0 : hb1);

        gemm16_kernel<<<gemmD, 128, 0, stream>>>(hin, Wq + (size_t)l * DD * DD,
                                                 bq + l * DD, nullptr, q, NN, DD, DD, 0);
        gemm16_kernel<<<gemmD, 128, 0, stream>>>(hin, Wk + (size_t)l * DD * DD,
                                                 bk + l * DD, nullptr, k, NN, DD, DD, 0);
        gemm16_kernel<<<gemmD, 128, 0, stream>>>(hin, Wv + (size_t)l * DD * DD,
                                                 bv + l * DD, nullptr, v, NN, DD, DD, 0);

        smax_init_kernel<<<(NN * HH + 255) / 256, 256, 0, stream>>>(smax);
        edge_smax_kernel<<<(EE * HH + 255) / 256, 256, 0, stream>>>(q, k, a_edge,
                                                                    src, dst, smax);

        attn_kernel<<<dim3(NN / 64, MCH), 128, 0, stream>>>(q, k, v, smax, ctx_part);
        ctx_reduce_kernel<<<(NN * DD + 255) / 256, 256, 0, stream>>>(ctx_part, ctx);

        // h2 = hin + ctx @ Wo + bo
        gemm16_kernel<<<gemmD, 128, 0, stream>>>(ctx, Wo + (size_t)l * DD * DD,
                                                 bo + l * DD, hin, h2, NN, DD, DD, 0);
        // ffn1 = relu(h2 @ W1 + b1)
        gemm16_kernel<<<gemmFF, 128, 0, stream>>>(h2, W1 + (size_t)l * DD * FFD,
                                                  b1 + l * FFD, nullptr, ffn1, NN, DD, FFD, 1);
        // z = h2 + ffn1 @ W2 + b2   (== h3)
        gemm16_kernel<<<gemmD, 128, 0, stream>>>(ffn1, W2 + (size_t)l * FFD * DD,
                                                 b2 + l * DD, h2, zbuf, NN, FFD, DD, 0);
        // hout = LN(hin + z)
        ln_kernel<<<NN, 128, 0, stream>>>(hin, zbuf, gam + l * DD, bet + l * DD, hout);
    }
}